// MLPPredictor_61495341744586
// MI455X (gfx1250) — compile-verified
//
#include <hip/hip_runtime.h>

// ---------------------------------------------------------------------------
// Per-edge score: score[e] = dot(user[src[e]], W[0:128])
//                          + dot(item[dst[e]], W[128:256]) + b
// E = 320000, d0 = d1 = 128, C = 1, all fp32.
//
// Memory-bound (0.5 FLOP/B); both feature tables (51 MB) are L2-resident on
// MI455X (192 MB L2), so the kernel is an L2-bandwidth gather. Compute path:
// V_WMMA_F32_16X16X4_F32, 16 edges per wave32, exact fp32 accumulation.
// ---------------------------------------------------------------------------

typedef float v2f __attribute__((ext_vector_type(2)));
typedef float v4f __attribute__((ext_vector_type(4)));
typedef float v8f __attribute__((ext_vector_type(8)));

#ifndef __has_builtin
#define __has_builtin(x) 0
#endif
#if __has_builtin(__builtin_amdgcn_wmma_f32_16x16x4_f32)
#define HAVE_WMMA_F32X4 1
#else
#define HAVE_WMMA_F32X4 0
#endif

#define WAVES_PER_BLOCK 8
#define BLOCK_THREADS (WAVES_PER_BLOCK * 32)
#define D0 128
#define D1 128

#if HAVE_WMMA_F32X4
// One wave32 computes 16 edge scores via 64x V_WMMA_F32_16X16X4_F32.
// A layout (16x4 f32): lanes 0-15 hold K=kb+0,kb+1; lanes 16-31 hold K=kb+2,kb+3
// for row M = lane%16.  B (4x16) gets the same W slice replicated in all 16
// columns, so every column of D equals the vector of 16 edge scores.
__global__ __launch_bounds__(BLOCK_THREADS)
void edge_score_wmma(const float* __restrict__ user_feats,
                     const float* __restrict__ item_feats,
                     const int*   __restrict__ src_idx,
                     const int*   __restrict__ dst_idx,
                     const float* __restrict__ W,
                     const float* __restrict__ bias_p,
                     float* __restrict__ out,
                     int n_tiles) {
  __shared__ __align__(16) float ldsW[D0 + D1];

  const int tid = threadIdx.x;
  if (tid < D0 + D1) ldsW[tid] = W[tid];
  __syncthreads();

  const int wave = tid >> 5;
  const int lane = tid & 31;
  const int tile = blockIdx.x * WAVES_PER_BLOCK + wave;
  if (tile >= n_tiles) return;  // wave-uniform: EXEC stays all-ones for WMMA

  const int m    = lane & 15;          // edge-in-tile served by this lane
  const int koff = (lane >> 4) << 1;   // 0 (lanes 0-15) or 2 (lanes 16-31)
  const int e    = tile * 16 + m;

  const float* rowU = user_feats + (size_t)src_idx[e] * D0;
  const float* rowI = item_feats + (size_t)dst_idx[e] * D1;
  __builtin_prefetch(rowU, 0, 3);
  __builtin_prefetch(rowI, 0, 3);

  v8f acc = {};
#pragma unroll 4
  for (int k = 0; k < D0; k += 4) {
    v2f a = *(const v2f*)(rowU + k + koff);
    v2f w = *(const v2f*)(ldsW + k + koff);
    acc = __builtin_amdgcn_wmma_f32_16x16x4_f32(false, a, false, w,
                                                (short)0, acc, false, false);
  }
#pragma unroll 4
  for (int k = 0; k < D1; k += 4) {
    v2f a = *(const v2f*)(rowI + k + koff);
    v2f w = *(const v2f*)(ldsW + D0 + k + koff);
    acc = __builtin_amdgcn_wmma_f32_16x16x4_f32(false, a, false, w,
                                                (short)0, acc, false, false);
  }

  // D VGPR r, lane l -> M = r + 8*(l>=16), N = l%16; all N columns identical.
  if (m == 0) {  // lanes 0 (M=0..7) and 16 (M=8..15)
    const float bias = bias_p[0];
    float* o = out + tile * 16 + ((lane >> 4) << 3);
    v4f lo = { acc[0] + bias, acc[1] + bias, acc[2] + bias, acc[3] + bias };
    v4f hi = { acc[4] + bias, acc[5] + bias, acc[6] + bias, acc[7] + bias };
    *(v4f*)(o)     = lo;
    *(v4f*)(o + 4) = hi;
  }
}
#endif  // HAVE_WMMA_F32X4

// Fallback / tail: one wave32 per edge. Each lane loads float4 of the user row
// and float4 of the item row (wave reads the full 512B row coalesced), FMAs
// with W, then a 5-step shuffle reduction.
__global__ __launch_bounds__(BLOCK_THREADS)
void edge_score_valu(const float* __restrict__ user_feats,
                     const float* __restrict__ item_feats,
                     const int*   __restrict__ src_idx,
                     const int*   __restrict__ dst_idx,
                     const float* __restrict__ W,
                     const float* __restrict__ bias_p,
                     float* __restrict__ out,
                     int e0, int count) {
  const int gwave = (int)((blockIdx.x * (unsigned)blockDim.x + threadIdx.x) >> 5);
  const int lane  = threadIdx.x & 31;
  if (gwave >= count) return;
  const int e = e0 + gwave;

  const float* rowU = user_feats + (size_t)src_idx[e] * D0;
  const float* rowI = item_feats + (size_t)dst_idx[e] * D1;

  v4f u  = *(const v4f*)(rowU + lane * 4);
  v4f wu = *(const v4f*)(W + lane * 4);
  v4f i  = *(const v4f*)(rowI + lane * 4);
  v4f wi = *(const v4f*)(W + D0 + lane * 4);

  float s = u[0] * wu[0] + u[1] * wu[1] + u[2] * wu[2] + u[3] * wu[3]
          + i[0] * wi[0] + i[1] * wi[1] + i[2] * wi[2] + i[3] * wi[3];

#pragma unroll
  for (int off = 16; off > 0; off >>= 1) s += __shfl_down(s, off, 32);

  if (lane == 0) out[e] = s + bias_p[0];
}

extern "C" void kernel_launch(void* const* d_in, const int* in_sizes, int n_in,
                              void* d_out, int out_size, void* d_ws, size_t ws_size,
                              hipStream_t stream) {
  const float* user_feats = (const float*)d_in[0];
  const float* item_feats = (const float*)d_in[1];
  const int*   src_idx    = (const int*)d_in[2];
  const int*   dst_idx    = (const int*)d_in[3];
  const float* W          = (const float*)d_in[4];
  const float* b          = (const float*)d_in[5];
  float* out = (float*)d_out;

  const int E       = in_sizes[2];
  const int n_tiles = E >> 4;       // 16 edges per wave tile
  const int rem     = E & 15;

#if HAVE_WMMA_F32X4
  if (n_tiles > 0) {
    int blocks = (n_tiles + WAVES_PER_BLOCK - 1) / WAVES_PER_BLOCK;
    edge_score_wmma<<<blocks, BLOCK_THREADS, 0, stream>>>(
        user_feats, item_feats, src_idx, dst_idx, W, b, out, n_tiles);
  }
#else
  if (n_tiles > 0) {
    int nwaves = n_tiles * 16;
    int blocks = (nwaves + WAVES_PER_BLOCK - 1) / WAVES_PER_BLOCK;
    edge_score_valu<<<blocks, BLOCK_THREADS, 0, stream>>>(
        user_feats, item_feats, src_idx, dst_idx, W, b, out, 0, nwaves);
  }
#endif
  if (rem > 0) {
    int blocks = (rem + WAVES_PER_BLOCK - 1) / WAVES_PER_BLOCK;
    edge_score_valu<<<blocks, BLOCK_THREADS, 0, stream>>>(
        user_feats, item_feats, src_idx, dst_idx, W, b, out, n_tiles * 16, rem);
  }
}